// DynamicMDETR_27608049778732
// MI455X (gfx1250) — compile-verified
//
#include <hip/hip_runtime.h>
#include <hip/hip_bf16.h>

// ---------------------------------------------------------------------------
// DynamicMDETR decoder for MI455X (gfx1250, wave32).
// Strategy: fp32 WMMA (V_WMMA_F32_16X16X4_F32) for all GEMMs (compute is far
// below roofline -> keep full fp32 precision), LDS-cached bilinear coords for
// the bandwidth-dominant grid-sample gather.
// ---------------------------------------------------------------------------

#define BATCH   1024
#define HID     256
#define PTS     36
#define FMW     20
#define FMHW    400      // 20*20
#define STAGES  6

typedef __attribute__((ext_vector_type(2))) float v2f;
typedef __attribute__((ext_vector_type(8))) float v8f;

// ---------------------------------------------------------------------------
// Generic WMMA GEMM: out[m][n] = act( sum_k A[m][k] * W[n][k] + bias[n] )
// A: (B x K) row-major.  W: (N x K) row-major (i.e. we compute A @ W^T).
// One wave (32 threads) computes a 16x16 output tile.
// grid.x = B/16, grid.y = ceil(N/16).  Nvalid handles N not multiple of 16.
//
// Fragment layouts per CDNA5 ISA 7.12.2:
//   A (16x4 f32, 2 VGPR): lanes 0-15 row=lane K={k0,k0+1}; lanes 16-31
//     row=lane-16 K={k0+2,k0+3}.
//   B (4x16 f32, 2 VGPR): B[k][n] = W[n][k]; same half-wave K split.
//   C/D (16x16 f32, 8 VGPR): vgpr e -> row e (lanes 0-15) / e+8 (lanes 16-31),
//     col = lane & 15.
// ---------------------------------------------------------------------------
__global__ __launch_bounds__(32) void wmma_gemm_f32(
    const float* __restrict__ A, const float* __restrict__ W,
    const float* __restrict__ bias, float* __restrict__ out,
    int K, int Nvalid, int do_relu)
{
    const int m0   = blockIdx.x << 4;
    const int n0   = blockIdx.y << 4;
    const int lane = threadIdx.x;          // 0..31
    const int half = lane >> 4;            // 0 or 1
    const int l    = lane & 15;

    const int arow = m0 + l;
    int ncol = n0 + l;
    if (ncol > Nvalid - 1) ncol = Nvalid - 1;   // clamp loads only (data, not control)

    const float* __restrict__ Ap = A + (size_t)arow * K + 2 * half;
    const float* __restrict__ Wp = W + (size_t)ncol * K + 2 * half;

    v8f c = {};
    // Uniform K-loop: EXEC stays all-ones across every WMMA issue.
    for (int k0 = 0; k0 < K; k0 += 4) {
        v2f a, b;
        a.x = Ap[k0];
        a.y = Ap[k0 + 1];
        b.x = Wp[k0];
        b.y = Wp[k0 + 1];
        // 8 args: (neg_a, A, neg_b, B, c_mod, C, reuse_a, reuse_b)
        c = __builtin_amdgcn_wmma_f32_16x16x4_f32(
                false, a, false, b, (short)0, c, false, false);
    }

    const int n = n0 + l;
    if (n < Nvalid) {
        const float bv = bias[n];
#pragma unroll
        for (int e = 0; e < 8; ++e) {
            const int m = m0 + e + 8 * half;
            float v = c[e] + bv;
            if (do_relu) v = fmaxf(v, 0.0f);
            out[(size_t)m * Nvalid + n] = v;
        }
    }
}

// ---------------------------------------------------------------------------
// Grid-sample (bilinear, border, align_corners=False) + mean pool over points,
// fused over (feature_map + pos) since bilinear is linear. Also materializes
// cat = [fused, q] for the update MLP.
// One block per batch, 256 threads = one per channel (8 waves).
// ---------------------------------------------------------------------------
__global__ __launch_bounds__(256) void sample_pool_cat(
    const float* __restrict__ xy,     // (B, 72)  raw offsets (x,y interleaved)
    const float* __restrict__ ref,    // (B, 2)
    const float* __restrict__ fm,     // (B, 256, 400)
    const float* __restrict__ pm,     // (B, 256, 400)
    const float* __restrict__ qbuf,   // (B, 256)
    float* __restrict__ cat)          // (B, 512): [:,0:256]=fused, [:,256:]=q
{
    const int b = blockIdx.x;
    const int c = threadIdx.x;

    __shared__ int   s_i00[PTS], s_i01[PTS], s_i10[PTS], s_i11[PTS];
    __shared__ float s_w00[PTS], s_w01[PTS], s_w10[PTS], s_w11[PTS];

    if (c < PTS) {
        // pts = xy + ref (in [0,1]); grid = 2*pts-1; unnormalize+border-clip
        // collapses to clamp(p*20 - 0.5, 0, 19).
        const float px = xy[b * 72 + 2 * c]     + ref[b * 2];
        const float py = xy[b * 72 + 2 * c + 1] + ref[b * 2 + 1];
        const float x  = fminf(fmaxf(px * 20.0f - 0.5f, 0.0f), 19.0f);
        const float y  = fminf(fmaxf(py * 20.0f - 0.5f, 0.0f), 19.0f);
        const float x0f = floorf(x), y0f = floorf(y);
        const int x0 = (int)x0f, y0 = (int)y0f;
        const int x1 = min(x0 + 1, FMW - 1);
        const int y1 = min(y0 + 1, FMW - 1);
        const float wx = x - x0f, wy = y - y0f;
        s_i00[c] = y0 * FMW + x0;  s_i01[c] = y0 * FMW + x1;
        s_i10[c] = y1 * FMW + x0;  s_i11[c] = y1 * FMW + x1;
        s_w00[c] = (1.0f - wx) * (1.0f - wy);
        s_w01[c] = wx * (1.0f - wy);
        s_w10[c] = (1.0f - wx) * wy;
        s_w11[c] = wx * wy;
    }
    __syncthreads();

    const size_t base = ((size_t)b * HID + c) * FMHW;
    const float* __restrict__ f = fm + base;
    const float* __restrict__ p = pm + base;

    float acc = 0.0f;
#pragma unroll 4
    for (int j = 0; j < PTS; ++j) {
        const int i00 = s_i00[j], i01 = s_i01[j], i10 = s_i10[j], i11 = s_i11[j];
        acc += s_w00[j] * (f[i00] + p[i00])
             + s_w01[j] * (f[i01] + p[i01])
             + s_w10[j] * (f[i10] + p[i10])
             + s_w11[j] * (f[i11] + p[i11]);
    }

    cat[(size_t)b * 512 + c]       = acc * (1.0f / (float)PTS);
    cat[(size_t)b * 512 + HID + c] = qbuf[(size_t)b * HID + c];
}

// ---------------------------------------------------------------------------
// Final bbox head: out[b][j] = sigmoid(dot(h[b], W3[j]) + b3[j]), j in [0,4).
// Tiny (4096 dots of length 256) -> scalar kernel.
// ---------------------------------------------------------------------------
__global__ __launch_bounds__(256) void bbox_head(
    const float* __restrict__ h, const float* __restrict__ W3,
    const float* __restrict__ b3, float* __restrict__ out)
{
    const int t = blockIdx.x * blockDim.x + threadIdx.x;
    if (t >= BATCH * 4) return;
    const int b = t >> 2;
    const int j = t & 3;
    const float* __restrict__ hr = h  + (size_t)b * HID;
    const float* __restrict__ wr = W3 + (size_t)j * HID;
    float acc = b3[j];
#pragma unroll 8
    for (int k = 0; k < HID; ++k) acc += hr[k] * wr[k];
    out[t] = 1.0f / (1.0f + __expf(-acc));
}

// ---------------------------------------------------------------------------
extern "C" void kernel_launch(void* const* d_in, const int* in_sizes, int n_in,
                              void* d_out, int out_size, void* d_ws, size_t ws_size,
                              hipStream_t stream) {
    (void)in_sizes; (void)n_in; (void)out_size; (void)ws_size;

    const float* sampling_query = (const float*)d_in[0];   // (B, 256)
    const float* reference_pt   = (const float*)d_in[1];   // (B, 2)
    const float* feature_map    = (const float*)d_in[2];   // (B, 256, 400)
    const float* pos            = (const float*)d_in[3];   // (B, 256, 400)
    const float* offset_W       = (const float*)d_in[4];   // (6, 72, 256)
    const float* offset_b       = (const float*)d_in[5];   // (6, 72)
    const float* up_W1          = (const float*)d_in[6];   // (6, 256, 512)
    const float* up_b1          = (const float*)d_in[7];   // (6, 256)
    const float* up_W2          = (const float*)d_in[8];   // (6, 256, 256)
    const float* up_b2          = (const float*)d_in[9];   // (6, 256)
    const float* bbox_W1        = (const float*)d_in[10];  // (256, 256)
    const float* bbox_b1        = (const float*)d_in[11];  // (256)
    const float* bbox_W2        = (const float*)d_in[12];  // (256, 256)
    const float* bbox_b2        = (const float*)d_in[13];  // (256)
    const float* bbox_W3        = (const float*)d_in[14];  // (4, 256)
    const float* bbox_b3        = (const float*)d_in[15];  // (4)
    float* out = (float*)d_out;                            // (B, 4)

    // Workspace layout (floats): q | xy | cat | hdd | h2   (~5.5 MB total)
    float* q   = (float*)d_ws;
    float* xy  = q   + (size_t)BATCH * HID;       // B*72 used
    float* cat = xy  + (size_t)BATCH * 72;        // B*512
    float* hdd = cat + (size_t)BATCH * 512;       // B*256
    float* h2  = hdd + (size_t)BATCH * HID;       // B*256

    // q <- sampling_query (inputs must not be mutated; q evolves in-place).
    hipMemcpyAsync(q, sampling_query, (size_t)BATCH * HID * sizeof(float),
                   hipMemcpyDeviceToDevice, stream);

    const dim3 wave(32);
    const dim3 gridOff(BATCH / 16, 5);    // N=72 -> 5 tiles (padded)
    const dim3 gridHid(BATCH / 16, HID / 16);

    for (int s = 0; s < STAGES; ++s) {
        // 1) offset generator: xy = q @ offset_W[s]^T + offset_b[s]   (B x 72)
        wmma_gemm_f32<<<gridOff, wave, 0, stream>>>(
            q, offset_W + (size_t)s * 72 * HID, offset_b + (size_t)s * 72,
            xy, HID, 72, 0);

        // 2) bilinear sample + mean pool + build cat = [fused, q]  (B x 512)
        sample_pool_cat<<<BATCH, 256, 0, stream>>>(
            xy, reference_pt, feature_map, pos, q, cat);

        // 3) update MLP layer 1: hdd = relu(cat @ up_W1[s]^T + up_b1[s])
        wmma_gemm_f32<<<gridHid, wave, 0, stream>>>(
            cat, up_W1 + (size_t)s * HID * 512, up_b1 + (size_t)s * HID,
            hdd, 512, HID, 1);

        // 4) update MLP layer 2: q = hdd @ up_W2[s]^T + up_b2[s] (in-place ok:
        //    each block reads/writes only its own 16 rows)
        wmma_gemm_f32<<<gridHid, wave, 0, stream>>>(
            hdd, up_W2 + (size_t)s * HID * HID, up_b2 + (size_t)s * HID,
            q, HID, HID, 0);
    }

    // bbox MLP: relu, relu, sigmoid
    wmma_gemm_f32<<<gridHid, wave, 0, stream>>>(q,   bbox_W1, bbox_b1, hdd, HID, HID, 1);
    wmma_gemm_f32<<<gridHid, wave, 0, stream>>>(hdd, bbox_W2, bbox_b2, h2,  HID, HID, 1);
    bbox_head<<<(BATCH * 4 + 255) / 256, 256, 0, stream>>>(h2, bbox_W3, bbox_b3, out);
}